// CRF_23424751633198
// MI455X (gfx1250) — compile-verified
//
#include <hip/hip_runtime.h>
#include <hip/hip_bf16.h>

#define NL     11
#define BOS    9
#define EOS    10
#define SEQ    8192
#define NFEAT  14
#define CHUNK  64
#define NCH    128          // 8191 steps -> 128 chunks of <=64
#define NPAIRS (SEQ * NL)   // 90112, divisible by 16

typedef __attribute__((ext_vector_type(2))) float v2f;
typedef __attribute__((ext_vector_type(8))) float v8f;

// ---------------------------------------------------------------------------
// Kernel 1: emissions[t*NL+l] = sum_f w[x[(t*NL+l)*14 + f]]
// Each wave32 handles 16 rows; 14 features padded to K=16 and reduced with
// four chained V_WMMA_F32_16X16X4_F32 against a ones B-matrix. Row sums land
// in column N=0 of D: lane 0 holds M=0..7 (VGPR 0..7), lane 16 holds M=8..15.
// Padding is branchless: pad lanes clamp their x-address to the last valid
// feature pair and get a zero B-column, so no EXEC-divergent load regions.
// ---------------------------------------------------------------------------
__global__ void crf_emissions_wmma(const int* __restrict__ x,
                                   const float* __restrict__ w,
                                   float* __restrict__ emis) {
    const int lane    = threadIdx.x & 31;
    const int wave    = (blockIdx.x * blockDim.x + threadIdx.x) >> 5;
    const int rowBase = wave * 16;
    if (rowBase >= NPAIRS) return;              // wave-uniform; EXEC stays full

    const int m     = lane & 15;                // A-matrix row this lane feeds
    const int khalf = lane >> 4;                // 0 -> K{0,1}, 1 -> K{2,3}
    const int row   = rowBase + m;
    const int* __restrict__ xrow = x + row * NFEAT;

    // Prefetch the x rows one wave-stride ahead (global_prefetch_b8).
    __builtin_prefetch(xrow + 16 * NFEAT, 0, 0);

    v8f acc = {};
#pragma unroll
    for (int kb = 0; kb < 16; kb += 4) {
        const int f0   = kb + 2 * khalf;        // features this lane feeds
        const bool pad = (f0 > NFEAT - 2);      // only kb=12, lanes 16-31
        const int fc   = pad ? (NFEAT - 2) : f0;  // clamp to valid pair
        float a0 = w[xrow[fc]];
        float a1 = w[xrow[fc + 1]];
        v2f a = {a0, a1};
        // Zero padded K-columns through B instead of masking loads:
        // D[m][n] = sum_k A[m][k]*B[k][n], so B=0 kills the pad contribution.
        float bv = pad ? 0.0f : 1.0f;
        v2f b = {bv, bv};
        acc = __builtin_amdgcn_wmma_f32_16x16x4_f32(
                  false, a, false, b, (short)0, acc, false, false);
    }
    // Column N=0 of D: lane 0 -> M=0..7, lane 16 -> M=8..15.
    if ((lane & 15) == 0) {
        const int mb = khalf * 8;
#pragma unroll
        for (int r = 0; r < 8; ++r) emis[rowBase + mb + r] = acc[r];
    }
}

// ---------------------------------------------------------------------------
// Kernel 2: per-chunk max-plus matrix product P_c = prod_{t in chunk} M_t,
// M_t[k][j] = T[k][j] + e_t[j].  One block (128 thr, 121 active) per chunk.
// ---------------------------------------------------------------------------
__global__ void crf_chunk_product(const float* __restrict__ emis,
                                  const float* __restrict__ trans,
                                  float* __restrict__ Pout) {
    __shared__ float P[NL * NL], Pn[NL * NL], T[NL * NL], e[NL];
    const int tid = threadIdx.x;
    const int c   = blockIdx.x;
    if (tid < NL * NL) {
        T[tid] = trans[tid];
        P[tid] = (tid / NL == tid % NL) ? 0.0f : -1.0e30f;   // tropical identity
    }
    __syncthreads();
    const int s0 = 1 + c * CHUNK;
    const int s1 = min(s0 + CHUNK, SEQ);        // steps t in [s0, s1)
    for (int t = s0; t < s1; ++t) {
        if (tid < NL) e[tid] = emis[t * NL + tid];
        __syncthreads();
        if (tid < NL * NL) {
            const int i = tid / NL, j = tid % NL;
            float best = -3.0e38f;
#pragma unroll
            for (int k = 0; k < NL; ++k)
                best = fmaxf(best, P[i * NL + k] + T[k * NL + j]);
            Pn[tid] = best + e[j];
        }
        __syncthreads();
        if (tid < NL * NL) P[tid] = Pn[tid];
        __syncthreads();
    }
    if (tid < NL * NL) Pout[c * NL * NL + tid] = P[tid];
}

// ---------------------------------------------------------------------------
// Kernel 3: propagate alpha through the 128 chunk products (serial but only
// 128 steps), record alpha at every chunk boundary, compute final score/tag.
// One block of 32 threads.
// ---------------------------------------------------------------------------
__global__ void crf_boundary_scan(const float* __restrict__ emis,
                                  const float* __restrict__ trans,
                                  const float* __restrict__ Pin,
                                  float* __restrict__ alphaStart,
                                  int* __restrict__ finalTag,
                                  float* __restrict__ d_out) {
    __shared__ float a[NL], an[NL];
    const int tid = threadIdx.x;
    if (tid < NL) a[tid] = trans[BOS * NL + tid] + emis[tid];   // alpha_0
    __syncthreads();
    for (int c = 0; c < NCH; ++c) {
        if (tid < NL) alphaStart[c * NL + tid] = a[tid];
        __syncthreads();
        if (tid < NL) {
            float best = -3.0e38f;
#pragma unroll
            for (int i = 0; i < NL; ++i)
                best = fmaxf(best, a[i] + Pin[c * NL * NL + i * NL + tid]);
            an[tid] = best;
        }
        __syncthreads();
        if (tid < NL) a[tid] = an[tid];
        __syncthreads();
    }
    if (tid == 0) {                              // end scores at t = 8191
        float best = -3.0e38f; int bi = 0;
#pragma unroll
        for (int i = 0; i < NL; ++i) {
            float s = a[i] + trans[i * NL + EOS];
            if (s > best) { best = s; bi = i; }  // first-max, matches argmax
        }
        d_out[0]   = best;
        *finalTag  = bi;
        d_out[SEQ] = (float)bi;                  // path position 8191
    }
}

// ---------------------------------------------------------------------------
// Kernel 4: replay each chunk from its boundary alpha, keep backpointers in
// LDS, then trace back for all 11 possible chunk-end tags to build the trace
// table tab[c][j][s] = tag at global position 64c+s.  One block (1 wave)/chunk.
// ---------------------------------------------------------------------------
__global__ void crf_replay_trace(const float* __restrict__ emis,
                                 const float* __restrict__ trans,
                                 const float* __restrict__ alphaStart,
                                 unsigned char* __restrict__ tab,
                                 int* __restrict__ startTag) {
    __shared__ float a[NL], an[NL], T[NL * NL];
    __shared__ unsigned char bp[CHUNK][NL];
    const int tid = threadIdx.x;
    const int c   = blockIdx.x;
    for (int idx = tid; idx < NL * NL; idx += 32) T[idx] = trans[idx];
    if (tid < NL) a[tid] = alphaStart[c * NL + tid];
    __syncthreads();
    const int s0  = 1 + c * CHUNK;
    const int len = min(s0 + CHUNK, SEQ) - s0;
    for (int s = 0; s < len; ++s) {
        const int t = s0 + s;
        if (tid < NL) {
            float best = -3.0e38f; int bi = 0;
#pragma unroll
            for (int i = 0; i < NL; ++i) {
                float sc = a[i] + T[i * NL + tid];
                if (sc > best) { best = sc; bi = i; }
            }
            an[tid]     = best + emis[t * NL + tid];
            bp[s][tid]  = (unsigned char)bi;
        }
        __syncthreads();
        if (tid < NL) a[tid] = an[tid];
        __syncthreads();
    }
    if (tid < NL) {                              // backtrace for end tag = tid
        int tag = tid;
        for (int s = len - 1; s >= 0; --s) {
            tag = bp[s][tag];
            tab[(c * NL + tid) * CHUNK + s] = (unsigned char)tag;
        }
        startTag[c * NL + tid] = tag;            // tag at position 64c
    }
}

// ---------------------------------------------------------------------------
// Kernel 5: 128-step serial stitch of chunk-end tags.
// ---------------------------------------------------------------------------
__global__ void crf_stitch(const int* __restrict__ startTag,
                           const int* __restrict__ finalTag,
                           int* __restrict__ endTag) {
    if (threadIdx.x == 0 && blockIdx.x == 0) {
        int tag = *finalTag;
        for (int c = NCH - 1; c >= 0; --c) {
            endTag[c] = tag;                     // tag at position e_c
            tag = startTag[c * NL + tag];        // tag at position 64c
        }
    }
}

// ---------------------------------------------------------------------------
// Kernel 6: parallel path emission for positions 0..8190.
// ---------------------------------------------------------------------------
__global__ void crf_emit_path(const unsigned char* __restrict__ tab,
                              const int* __restrict__ endTag,
                              float* __restrict__ d_out) {
    const int t = blockIdx.x * blockDim.x + threadIdx.x;
    if (t < SEQ - 1) {
        const int c = t >> 6, s = t & 63;
        d_out[1 + t] = (float)tab[(c * NL + endTag[c]) * CHUNK + s];
    }
}

extern "C" void kernel_launch(void* const* d_in, const int* in_sizes, int n_in,
                              void* d_out, int out_size, void* d_ws, size_t ws_size,
                              hipStream_t stream) {
    const int*   x     = (const int*)d_in[0];     // (8192, 11, 14) int32
    const float* w     = (const float*)d_in[1];   // (20M, 1) f32
    const float* trans = (const float*)d_in[2];   // (11, 11) f32
    float* out = (float*)d_out;

    // Workspace layout (16B-aligned offsets), total < 600 KB.
    char* ws = (char*)d_ws;
    size_t off = 0;
    auto alloc = [&](size_t bytes) { void* p = ws + off; off = (off + bytes + 15) & ~size_t(15); return p; };
    float*         emis       = (float*)alloc(sizeof(float) * NPAIRS);
    float*         P          = (float*)alloc(sizeof(float) * NCH * NL * NL);
    float*         alphaStart = (float*)alloc(sizeof(float) * NCH * NL);
    unsigned char* tab        = (unsigned char*)alloc((size_t)NCH * NL * CHUNK);
    int*           startTag   = (int*)alloc(sizeof(int) * NCH * NL);
    int*           finalTag   = (int*)alloc(sizeof(int));
    int*           endTag     = (int*)alloc(sizeof(int) * NCH);

    // 1) emissions: 5632 waves of 16 rows each -> 704 blocks x 256 threads
    crf_emissions_wmma<<<NPAIRS / 128, 256, 0, stream>>>(x, w, emis);
    // 2) per-chunk tropical matrix products
    crf_chunk_product<<<NCH, 128, 0, stream>>>(emis, trans, P);
    // 3) boundary alpha scan + final score/tag
    crf_boundary_scan<<<1, 32, 0, stream>>>(emis, trans, P, alphaStart, finalTag, out);
    // 4) per-chunk replay + trace tables
    crf_replay_trace<<<NCH, 32, 0, stream>>>(emis, trans, alphaStart, tab, startTag);
    // 5) stitch chunk-end tags (128 serial steps)
    crf_stitch<<<1, 32, 0, stream>>>(startTag, finalTag, endTag);
    // 6) emit best path in parallel
    crf_emit_path<<<(SEQ - 1 + 255) / 256, 256, 0, stream>>>(tab, endTag, out);
}